// MultiBoxLoss_8083128451238
// MI455X (gfx1250) — compile-verified
//
#include <hip/hip_runtime.h>
#include <hip/hip_bf16.h>

// ---- problem constants (fixed by reference setup) ----
#define Bc   64
#define Dc   8732
#define Cc   81
#define Nc   50
#define TPB  256
#define NBX  ((Dc + TPB - 1) / TPB)   // 35 blocks along D
#define NBLK (Bc * NBX)               // 2240 partial slots (multiple of 64)

typedef __attribute__((ext_vector_type(2))) float v2f;
typedef __attribute__((ext_vector_type(8))) float v8f;

// =====================================================================
// Stage 1: fused match + offset-encode + smooth-L1 + log-softmax CE.
// One thread per (b, d). Reads conf_data exactly once (the 181MB stream
// that bounds this op at ~8us @ 23.3 TB/s).
// =====================================================================
__global__ void ssd_match_ce_kernel(const float* __restrict__ loc,
                                    const float* __restrict__ conf,
                                    const float* __restrict__ dboxes,
                                    const float* __restrict__ tb,
                                    const int*   __restrict__ tl,
                                    float* __restrict__ ce_mined,
                                    float* __restrict__ part_l,
                                    float* __restrict__ part_ce,
                                    int*   __restrict__ npos)
{
    __shared__ float s_tb[Nc * 4];
    __shared__ float s_area[Nc];
    __shared__ int   s_tl[Nc];
    __shared__ float sh_l[TPB];
    __shared__ float sh_c[TPB];
    __shared__ int   sh_n[TPB];

    const int b   = blockIdx.y;
    const int tid = threadIdx.x;

    for (int i = tid; i < Nc * 4; i += TPB) s_tb[i] = tb[(size_t)b * Nc * 4 + i];
    for (int i = tid; i < Nc;     i += TPB) s_tl[i] = tl[b * Nc + i];
    __syncthreads();
    if (tid < Nc) {
        float x0 = s_tb[tid*4+0], y0 = s_tb[tid*4+1];
        float x1 = s_tb[tid*4+2], y1 = s_tb[tid*4+3];
        s_area[tid] = (x0 - x1) * (y0 - y1);   // matches reference _area (positive)
    }
    __syncthreads();

    const int d = blockIdx.x * TPB + tid;
    float sl1_sum = 0.0f;
    float ce_pos  = 0.0f;
    int   is_pos  = 0;

    if (d < Dc) {
        const float dx0 = dboxes[d*4+0], dy0 = dboxes[d*4+1];
        const float dx1 = dboxes[d*4+2], dy1 = dboxes[d*4+3];
        const float area_d = (dx0 - dx1) * (dy0 - dy1);

        float best_iou = -1.0f;
        int   best     = 0;
        #pragma unroll 5
        for (int n = 0; n < Nc; ++n) {
            float tx0 = s_tb[n*4+0], ty0 = s_tb[n*4+1];
            float tx1 = s_tb[n*4+2], ty1 = s_tb[n*4+3];
            float lx = fmaxf(tx0, dx0), ly = fmaxf(ty0, dy0);
            float rx = fminf(tx1, dx1), ry = fminf(ty1, dy1);
            float w  = fmaxf(rx - lx, 0.0f), h = fmaxf(ry - ly, 0.0f);
            float inter = w * h;
            float iou = inter / (s_area[n] + area_d - inter);
            if (iou > best_iou) { best_iou = iou; best = n; }   // first-max == argmax
        }
        int conf_t = (best_iou < 0.5f) ? 0 : (s_tl[best] + 1);
        is_pos = (conf_t > 0) ? 1 : 0;

        // --- per-anchor CE from log-softmax over 81 classes ---
        const float* row = conf + ((size_t)b * Dc + d) * Cc;
        float m = row[0];
        for (int c = 1; c < Cc; ++c) m = fmaxf(m, row[c]);
        float s = 0.0f;
        for (int c = 0; c < Cc; ++c) s += expf(row[c] - m);
        float ce = m + logf(s) - row[conf_t];

        ce_mined[(size_t)b * Dc + d] = is_pos ? 0.0f : ce;   // ce >= 0 always

        if (is_pos) {
            ce_pos = ce;
            // --- SSD variance-scaled offset encode + smooth-L1 ---
            float bx0 = s_tb[best*4+0], by0 = s_tb[best*4+1];
            float bx1 = s_tb[best*4+2], by1 = s_tb[best*4+3];
            float dsw = dx1 - dx0, dsh = dy1 - dy0;
            float tsw = bx1 - bx0, tsh = by1 - by0;
            float g0 = ((bx0 - dx0) + (bx1 - dx1)) * 0.5f * 10.0f / dsw;
            float g1 = ((by0 - dy0) + (by1 - dy1)) * 0.5f * 10.0f / dsh;
            float g2 = logf(tsw / dsw) * 5.0f;
            float g3 = logf(tsh / dsh) * 5.0f;
            float g[4] = {g0, g1, g2, g3};
            const float* lp = loc + ((size_t)b * Dc + d) * 4;
            #pragma unroll
            for (int i = 0; i < 4; ++i) {
                float x  = lp[i] - g[i];
                float ax = fabsf(x);
                sl1_sum += (ax < 1.0f) ? 0.5f * x * x : (ax - 0.5f);
            }
        }
    }

    // --- block reduction (deterministic tree), fixed partial slots ---
    sh_l[tid] = sl1_sum; sh_c[tid] = ce_pos; sh_n[tid] = is_pos;
    __syncthreads();
    for (int off = TPB / 2; off > 0; off >>= 1) {
        if (tid < off) {
            sh_l[tid] += sh_l[tid + off];
            sh_c[tid] += sh_c[tid + off];
            sh_n[tid] += sh_n[tid + off];
        }
        __syncthreads();
    }
    if (tid == 0) {
        const int slot = b * NBX + blockIdx.x;
        part_l[slot]  = sh_l[0];
        part_ce[slot] = sh_c[0];
        atomicAdd(&npos[b], sh_n[0]);     // integer atomic: deterministic
    }
}

// =====================================================================
// Stage 2: per-batch top-k sum of ce_mined via 31-step radix search on
// float bit patterns (all values >= 0 => uint order == float order).
// Tie-exact vs the reference's double-argsort selection.
// =====================================================================
__global__ void topk_sum_kernel(const float* __restrict__ ce_mined,
                                const int*   __restrict__ npos,
                                float* __restrict__ topk)
{
    __shared__ int   cnt_s[TPB];
    __shared__ float sum_s[TPB];
    const int b   = blockIdx.x;
    const int tid = threadIdx.x;
    const float* v = ce_mined + (size_t)b * Dc;

    const int np = npos[b];
    int k = np * 3; if (k > Dc) k = Dc;
    if (k == 0) { if (tid == 0) topk[b] = 0.0f; return; }

    unsigned t = 0u;
    for (int bit = 30; bit >= 0; --bit) {          // sign bit is always 0
        unsigned cand = t | (1u << bit);
        float cf = __uint_as_float(cand);
        int c = 0;
        for (int i = tid; i < Dc; i += TPB) c += (v[i] >= cf) ? 1 : 0;
        cnt_s[tid] = c; __syncthreads();
        for (int off = TPB / 2; off > 0; off >>= 1) {
            if (tid < off) cnt_s[tid] += cnt_s[tid + off];
            __syncthreads();
        }
        int total = cnt_s[0];
        __syncthreads();
        if (total >= k) t = cand;                  // uniform across block
    }
    const float tf = __uint_as_float(t);           // k-th largest value

    int cg = 0; float sg = 0.0f;
    for (int i = tid; i < Dc; i += TPB) {
        float x = v[i];
        if (x > tf) { ++cg; sg += x; }
    }
    cnt_s[tid] = cg; sum_s[tid] = sg; __syncthreads();
    for (int off = TPB / 2; off > 0; off >>= 1) {
        if (tid < off) { cnt_s[tid] += cnt_s[tid + off]; sum_s[tid] += sum_s[tid + off]; }
        __syncthreads();
    }
    if (tid == 0) topk[b] = sum_s[0] + (float)(k - cnt_s[0]) * tf;
}

// =====================================================================
// Stage 3: single-wave finalize. Partial sums reduced with
// V_WMMA_F32_16X16X4_F32: B = ones => D = A x 1 + C accumulates row sums
// of 64 values per issue, full f32 precision. EXEC is all-ones (32 thr).
// =====================================================================
__device__ __forceinline__ float wmma_sum64(const float* __restrict__ p, int groups)
{
    v8f acc = {};
    v2f bone; bone[0] = 1.0f; bone[1] = 1.0f;
    const int lane = threadIdx.x;                  // 0..31, one wave32
    for (int g = 0; g < groups; ++g) {
        v2f a;
        a[0] = p[g * 64 + lane * 2 + 0];
        a[1] = p[g * 64 + lane * 2 + 1];
        // 8 args: (neg_a, A, neg_b, B, c_mod, C, reuse_a, reuse_b)
        acc = __builtin_amdgcn_wmma_f32_16x16x4_f32(
            false, a, false, bone, (short)0, acc, false, false);
    }
    // column 0 of D lives in lane 0 (M=0..7) and lane 16 (M=8..15)
    float col = acc[0]+acc[1]+acc[2]+acc[3]+acc[4]+acc[5]+acc[6]+acc[7];
    float c0  = __shfl(col, 0, 32);
    float c16 = __shfl(col, 16, 32);
    return c0 + c16;
}

__global__ void finalize_kernel(const float* __restrict__ part_l,
                                const float* __restrict__ part_ce,
                                const float* __restrict__ topk,
                                const int*   __restrict__ npos,
                                float* __restrict__ out)
{
    const float L  = wmma_sum64(part_l,  NBLK / 64);  // 2240 = 35*64
    const float CE = wmma_sum64(part_ce, NBLK / 64);
    const float TK = wmma_sum64(topk,    1);          // 64 per-batch top-k sums

    const int lane = threadIdx.x;
    int n = npos[lane * 2] + npos[lane * 2 + 1];
    for (int off = 16; off > 0; off >>= 1) n += __shfl_down(n, off, 32);
    n = __shfl(n, 0, 32);

    if (lane == 0) {
        float Nt = (float)n;
        out[0] = L / Nt;
        out[1] = (CE + TK) / Nt;
    }
}

// =====================================================================
extern "C" void kernel_launch(void* const* d_in, const int* in_sizes, int n_in,
                              void* d_out, int out_size, void* d_ws, size_t ws_size,
                              hipStream_t stream)
{
    const float* loc  = (const float*)d_in[0];   // [B,D,4]
    const float* conf = (const float*)d_in[1];   // [B,D,C]
    const float* dbx  = (const float*)d_in[2];   // [D,4]
    const float* tbx  = (const float*)d_in[3];   // [B,N,4]
    const int*   tlb  = (const int*)  d_in[4];   // [B,N]

    // workspace layout (~2.26 MB; L2-resident)
    float* ce      = (float*)d_ws;               // B*D
    float* part_l  = ce + (size_t)Bc * Dc;       // NBLK
    float* part_ce = part_l + NBLK;              // NBLK
    float* topk    = part_ce + NBLK;             // Bc
    int*   npos    = (int*)(topk + Bc);          // Bc

    hipMemsetAsync(npos, 0, Bc * sizeof(int), stream);

    dim3 g1(NBX, Bc);
    ssd_match_ce_kernel<<<g1, TPB, 0, stream>>>(loc, conf, dbx, tbx, tlb,
                                                ce, part_l, part_ce, npos);
    topk_sum_kernel<<<Bc, TPB, 0, stream>>>(ce, npos, topk);
    finalize_kernel<<<1, 32, 0, stream>>>(part_l, part_ce, topk, npos, (float*)d_out);
}